// ExhaustiveBiaffineNERDecoder_2345052143875
// MI455X (gfx1250) — compile-verified
//
#include <hip/hip_runtime.h>
#include <hip/hip_bf16.h>

typedef float v2f __attribute__((ext_vector_type(2)));
typedef float v4f __attribute__((ext_vector_type(4)));
typedef float v8f __attribute__((ext_vector_type(8)));
typedef int   v4i __attribute__((ext_vector_type(4)));

#define BN_EPS 1e-5f

constexpr int kB  = 8;
constexpr int kT  = 512;
constexpr int kD  = 768;
constexpr int kBT = kB * kT;          // 4096 tokens
constexpr int kL  = 16;               // N_LABELS
constexpr int kLD = 128;              // LABEL_DIM
constexpr long kSC = (long)kB * kL * kT * kT;   // 33,554,432 score elements

// workspace layout in floats
constexpr int  WS_SCALE = 0;
constexpr int  WS_SHIFT = 1024;
constexpr int  WS_HS    = 4096;                         // [B,L,T,LD] start features
constexpr int  WS_HE    = WS_HS + kB * kL * kT * kLD;   // [B,L,T,LD] end features

constexpr int KC   = 32;            // K-chunk staged in LDS
constexpr int LDA  = 36;            // padded LDS row stride (gcd(36,64)=4 -> conflict-free)
constexpr int TILE = 128 * LDA;     // floats per tile buffer

#if defined(__gfx1250__) && __has_builtin(__builtin_amdgcn_global_load_async_to_lds_b128) && \
    __has_builtin(__builtin_amdgcn_s_wait_asynccnt)
#define USE_ASYNC 1
#else
#define USE_ASYNC 0
#endif

#if USE_ASYNC
typedef __attribute__((address_space(1))) v4i* gptr4;   // global
typedef __attribute__((address_space(3))) v4i* lptr4;   // LDS
#endif

// 16-byte global -> LDS copy (async on CDNA5 if available)
__device__ __forceinline__ void copy16_to_lds(const float* g, float* l) {
#if USE_ASYNC
  __builtin_amdgcn_global_load_async_to_lds_b128((gptr4)g, (lptr4)l, 0, 0);
#else
  *(v4f*)l = *(const v4f*)g;
#endif
}

__device__ __forceinline__ void async_join() {
#if USE_ASYNC
  __builtin_amdgcn_s_wait_asynccnt(0);
#endif
}

// ---------------------------------------------------------------------------
// Kernel 1: masked BatchNorm statistics -> fused scale/shift per feature
// ---------------------------------------------------------------------------
__global__ __launch_bounds__(256) void bn_stats_kernel(
    const float* __restrict__ x, const unsigned char* __restrict__ mask,
    const float* __restrict__ gamma, const float* __restrict__ beta,
    float* __restrict__ ws) {
  int d = blockIdx.x * blockDim.x + threadIdx.x;
  if (d >= kD) return;
  float s = 0.f, sq = 0.f;
  int cnt = 0;
  for (int r = 0; r < kBT; ++r) {
    float m = mask[r] ? 1.f : 0.f;
    float v = x[r * kD + d];
    s  += v * m;
    sq += v * v * m;
    cnt += mask[r] ? 1 : 0;
  }
  float denom = fmaxf((float)cnt, 1.f);
  float mean  = s / denom;
  float var   = sq / denom - mean * mean;
  float sc    = gamma[d] * rsqrtf(var + BN_EPS);
  ws[WS_SCALE + d] = sc;
  ws[WS_SHIFT + d] = beta[d] - mean * sc;
}

// ---------------------------------------------------------------------------
// Staging helpers (128 x 32 tiles, 4 float4 per thread)
// ---------------------------------------------------------------------------
__device__ __forceinline__ void g1_stage_a(
    float* __restrict__ dst, const float* __restrict__ x,
    const float* __restrict__ scale, const float* __restrict__ shift,
    const unsigned char* __restrict__ mask, int row0, int kc, int tid) {
#pragma unroll
  for (int it = 0; it < 4; ++it) {
    int idx4 = it * 256 + tid;        // 0..1023 float4 slots
    int i    = idx4 >> 3;             // row 0..127
    int j4   = (idx4 & 7) << 2;       // col 0,4,...,28
    int gr   = row0 + i;
    int dd   = kc + j4;
    v4f xv = *(const v4f*)(x + (long)gr * kD + dd);
    v4f sc = *(const v4f*)(scale + dd);
    v4f sh = *(const v4f*)(shift + dd);
    float m = mask[gr] ? 1.f : 0.f;
    v4f av;
    av.x = (xv.x * sc.x + sh.x) * m;
    av.y = (xv.y * sc.y + sh.y) * m;
    av.z = (xv.z * sc.z + sh.z) * m;
    av.w = (xv.w * sc.w + sh.w) * m;
    *(v4f*)(&dst[i * LDA + j4]) = av;
  }
}

__device__ __forceinline__ void g1_stage_b(
    float* __restrict__ dst, const float* __restrict__ w,
    int col0, int kc, int tid) {
#pragma unroll
  for (int it = 0; it < 4; ++it) {
    int idx4 = it * 256 + tid;
    int i    = idx4 >> 3;
    int j4   = (idx4 & 7) << 2;
    copy16_to_lds(w + (long)(col0 + i) * kD + kc + j4, &dst[i * LDA + j4]);
  }
}

__device__ __forceinline__ void g2_stage(
    float* __restrict__ dstS, float* __restrict__ dstE,
    const float* __restrict__ Sp, const float* __restrict__ Ep,
    int kc, int tid) {
#pragma unroll
  for (int it = 0; it < 4; ++it) {
    int idx4 = it * 256 + tid;
    int i    = idx4 >> 3;
    int j4   = (idx4 & 7) << 2;
    copy16_to_lds(Sp + (long)i * kLD + kc + j4, &dstS[i * LDA + j4]);
    copy16_to_lds(Ep + (long)i * kLD + kc + j4, &dstE[i * LDA + j4]);
  }
}

// ---------------------------------------------------------------------------
// Kernel 2: FFN GEMM  [4096 tokens, 768] x [768, 4096] + bias + ReLU,
// de-interleaved into Hs / He workspaces. f32 WMMA 16x16x4, double-buffered.
// ---------------------------------------------------------------------------
__global__ __launch_bounds__(256) void gemm1_kernel(
    const float* __restrict__ x, const unsigned char* __restrict__ mask,
    const float* __restrict__ w, const float* __restrict__ bias,
    float* __restrict__ ws) {
  __shared__ float As[2 * TILE];
  __shared__ float Bs[2 * TILE];

  const float* scale = ws + WS_SCALE;
  const float* shift = ws + WS_SHIFT;
  float* Hs = ws + WS_HS;
  float* He = ws + WS_HE;

  const int br   = (int)blockIdx.x >> 5;   // token-row block 0..31
  const int bc   = (int)blockIdx.x & 31;   // output-col block 0..31
  const int tid  = threadIdx.x;
  const int lane = tid & 31;
  const int wv   = tid >> 5;               // wave 0..7
  const int hl   = lane >> 4;              // half-lane 0/1
  const int ln   = lane & 15;

  const int row0 = br * 128;
  const int col0 = bc * 128;

  v8f acc[8];
#pragma unroll
  for (int j = 0; j < 8; ++j)
#pragma unroll
    for (int r = 0; r < 8; ++r) acc[j][r] = 0.f;

  constexpr int NC = kD / KC;   // 24 chunks
  g1_stage_b(Bs, w, col0, 0, tid);     // async copies (or manual fallback)
  g1_stage_a(As, x, scale, shift, mask, row0, 0, tid);

  int cur = 0;
  for (int c = 0; c < NC; ++c) {
    async_join();
    __syncthreads();
    if (c + 1 < NC) {
      int kn = (c + 1) * KC;
      g1_stage_b(Bs + (cur ^ 1) * TILE, w, col0, kn, tid);
      g1_stage_a(As + (cur ^ 1) * TILE, x, scale, shift, mask, row0, kn, tid);
    }
    const float* Ab = As + cur * TILE;
    const float* Bb = Bs + cur * TILE;
#pragma unroll
    for (int ks = 0; ks < KC; ks += 4) {
      int kb = ks + (hl << 1);
      v2f a = *(const v2f*)(&Ab[(wv * 16 + ln) * LDA + kb]);
#pragma unroll
      for (int j = 0; j < 8; ++j) {
        v2f b = *(const v2f*)(&Bb[(j * 16 + ln) * LDA + kb]);
        acc[j] = __builtin_amdgcn_wmma_f32_16x16x4_f32(
            false, a, false, b, (short)0, acc[j], false, false);
      }
    }
    cur ^= 1;
  }

  // Epilogue: bias + ReLU, de-interleave even/odd outputs into Hs/He.
#pragma unroll
  for (int j = 0; j < 8; ++j) {
    int o  = col0 + j * 16 + ln;
    float bb = bias[o];
    int l  = o >> 8;            // label
    int dd = (o >> 1) & 127;    // label_dim index
    float* Hp = (o & 1) ? He : Hs;
#pragma unroll
    for (int r = 0; r < 8; ++r) {
      int m  = r + (hl << 3);
      int gr = row0 + wv * 16 + m;
      int b_ = gr >> 9;          // batch
      int t  = gr & 511;         // token
      float v = acc[j][r] + bb;
      v = v > 0.f ? v : 0.f;
      Hp[(((long)(b_ * kL + l) * kT + t) * kLD) + dd] = v;
    }
  }
}

// ---------------------------------------------------------------------------
// Kernel 3: biaffine scores per (b,l): Hs[512,128] x He[512,128]^T + bias,
// plus triu & mask span-mask. Grid: 128 pairs x 16 sub-blocks of 128x128.
// ---------------------------------------------------------------------------
__global__ __launch_bounds__(256) void gemm2_kernel(
    const float* __restrict__ ws, const unsigned char* __restrict__ mask,
    const float* __restrict__ label_bias, float* __restrict__ out) {
  __shared__ float Ss[2 * TILE];
  __shared__ float Es[2 * TILE];

  const float* Hs = ws + WS_HS;
  const float* He = ws + WS_HE;

  const int blk  = blockIdx.x;
  const int pair = blk >> 4;        // b*16 + l, 0..127
  const int rem  = blk & 15;
  const int sb   = rem >> 2;        // start-row block 0..3
  const int eb   = rem & 3;         // end-col block 0..3
  const int b_   = pair >> 4;
  const int l    = pair & 15;

  const int tid  = threadIdx.x;
  const int lane = tid & 31;
  const int wv   = tid >> 5;
  const int hl   = lane >> 4;
  const int ln   = lane & 15;

  const float* Sp = Hs + (long)pair * kT * kLD + (long)sb * 128 * kLD;
  const float* Ep = He + (long)pair * kT * kLD + (long)eb * 128 * kLD;

  v8f acc[8];
#pragma unroll
  for (int j = 0; j < 8; ++j)
#pragma unroll
    for (int r = 0; r < 8; ++r) acc[j][r] = 0.f;

  constexpr int NC = kLD / KC;   // 4 chunks
  g2_stage(Ss, Es, Sp, Ep, 0, tid);

  int cur = 0;
  for (int c = 0; c < NC; ++c) {
    async_join();
    __syncthreads();
    if (c + 1 < NC) {
      g2_stage(Ss + (cur ^ 1) * TILE, Es + (cur ^ 1) * TILE, Sp, Ep, (c + 1) * KC, tid);
    }
    const float* Sb = Ss + cur * TILE;
    const float* Eb = Es + cur * TILE;
#pragma unroll
    for (int ks = 0; ks < KC; ks += 4) {
      int kb = ks + (hl << 1);
      v2f a = *(const v2f*)(&Sb[(wv * 16 + ln) * LDA + kb]);
#pragma unroll
      for (int j = 0; j < 8; ++j) {
        v2f b = *(const v2f*)(&Eb[(j * 16 + ln) * LDA + kb]);
        acc[j] = __builtin_amdgcn_wmma_f32_16x16x4_f32(
            false, a, false, b, (short)0, acc[j], false, false);
      }
    }
    cur ^= 1;
  }

  const float lb = label_bias[l];
  float* scores = out;
  float* smask  = out + kSC;

#pragma unroll
  for (int j = 0; j < 8; ++j) {
    int e = eb * 128 + j * 16 + ln;
    bool me = mask[b_ * kT + e] != 0;
#pragma unroll
    for (int r = 0; r < 8; ++r) {
      int m = r + (hl << 3);
      int s = sb * 128 + wv * 16 + m;
      long off = ((long)pair * kT + s) * kT + e;
      scores[off] = acc[j][r] + lb;
      bool ok = (s <= e) && (mask[b_ * kT + s] != 0) && me;
      smask[off] = ok ? 1.f : 0.f;
    }
  }
}

// ---------------------------------------------------------------------------
extern "C" void kernel_launch(void* const* d_in, const int* in_sizes, int n_in,
                              void* d_out, int out_size, void* d_ws, size_t ws_size,
                              hipStream_t stream) {
  const float*         feats = (const float*)d_in[0];
  const unsigned char* mask  = (const unsigned char*)d_in[1];  // numpy bool layout
  const float*         gamma = (const float*)d_in[2];
  const float*         beta  = (const float*)d_in[3];
  const float*         ffw   = (const float*)d_in[4];
  const float*         ffb   = (const float*)d_in[5];
  const float*         lbias = (const float*)d_in[6];
  float* wsf = (float*)d_ws;
  float* out = (float*)d_out;

  hipLaunchKernelGGL(bn_stats_kernel, dim3(3), dim3(256), 0, stream,
                     feats, mask, gamma, beta, wsf);
  hipLaunchKernelGGL(gemm1_kernel, dim3(1024), dim3(256), 0, stream,
                     feats, mask, ffw, ffb, wsf);
  hipLaunchKernelGGL(gemm2_kernel, dim3(2048), dim3(256), 0, stream,
                     wsf, mask, lbias, out);
}